// Pycontourlet_77824807404018
// MI455X (gfx1250) — compile-verified
//
#include <hip/hip_runtime.h>
#include <hip/hip_bf16.h>

typedef __attribute__((ext_vector_type(2))) float v2f;
typedef __attribute__((ext_vector_type(4))) float v4f;
typedef __attribute__((ext_vector_type(8))) float v8f;

// maxflat analysis filters (values fold the M*sqrt(2)==1 normalization)
__device__ __forceinline__ float tap_h(int t) {
    float v = 0.0f;
    v = (t == 0) ? 0.125f               : v;
    v = (t == 1) ? 0.35355339059327373f : v;
    v = (t == 2) ? 1.25f                : v;
    v = (t == 3) ? 0.35355339059327373f : v;
    v = (t == 4) ? 0.125f               : v;
    return v;
}
__device__ __forceinline__ float tap_h1(int t) {
    float v = 0.0f;
    v = (t == 0) ?  0.025888347648318447f : v;
    v = (t == 1) ?  0.07322330470336313f  : v;
    v = (t == 2) ? -0.06878156646177088f  : v;
    v = (t == 3) ? -0.8535533905932737f   : v;
    v = (t == 4) ? -0.06878156646177088f  : v;
    v = (t == 5) ?  0.07322330470336313f  : v;
    v = (t == 6) ?  0.025888347648318447f : v;
    return v;
}

#define PIN 76   // LDS pitch of input tile (80 x 76), mult of 4 for b128 stores
#define PMI 40   // LDS pitch of intermediates (80 x 40)

// One workgroup -> 32x32 outputs of each of the 4 subbands for one image.
// Row stage:  Y[m][n] = sum_p  Xin[m][p]   * F[p][n]   (A=data, B=filter)
// Col stage:  Y[i][n] = sum_p  F[i-row][p] * Xi[p][n]  (A=filter, B=data)
// F[p][n] = f[p - 2n - 4 + e], zero outside [0,L). K padded to 40 -> 10 chunks.
__global__ void __launch_bounds__(256)
wfb2dec_wmma_kernel(const float* __restrict__ xin, float* __restrict__ out)
{
    __shared__ __align__(16) float s_in[80 * PIN];      // input tile (wrapped)
    __shared__ __align__(16) float s_xl[80 * PMI];      // row-filtered w/ h
    __shared__ __align__(16) float s_xh[80 * PMI];      // row-filtered w/ h1
    __shared__ __align__(16) float s_fm[2 * 10 * 64];   // filter matrices (B/A layout)

    const int tid  = threadIdx.x;
    // wave id as an SGPR: task loops become scalar-branch controlled, so EXEC
    // is provably all-ones at every WMMA (ISA requirement).
    const int wave = __builtin_amdgcn_readfirstlane(tid) >> 5;
    const int lane = tid & 31;
    const int lm   = lane & 15;   // lane % 16
    const int lh   = lane >> 4;   // lane / 16

    const int ox0 = blockIdx.x * 32;
    const int oy0 = blockIdx.y * 32;
    const int img = blockIdx.z;

    const int col0 = 2 * ox0 - 4 + 1024;  // 4-aligned; +1024 so & 1023 wraps
    const int row0 = 2 * oy0 - 4 + 1024;
    const float* src = xin + (size_t)img * (1024u * 1024u);

    // --- build filter matrices once (values depend only on lane id) ---
    if (tid < 32) {
        #pragma unroll
        for (int ff = 0; ff < 2; ++ff) {
            const int e = ff ? 4 : 2;
            #pragma unroll
            for (int c = 0; c < 10; ++c) {
                #pragma unroll
                for (int v = 0; v < 2; ++v) {
                    const int p = c * 4 + 2 * lh + v;   // K coordinate
                    const int t = p - 2 * lm - 4 + e;   // filter tap index
                    s_fm[ff * 640 + c * 64 + lane * 2 + v] = ff ? tap_h1(t) : tap_h(t);
                }
            }
        }
    }

    // --- global -> LDS, b128, periodic wrap on both axes ---
    // col0 and the wrap point (1024) are both 4-aligned, so a 4-column group
    // never straddles the wrap: one wrap-safe float4 load per group.
    for (int idx = tid; idx < 80 * (PIN / 4); idx += 256) {
        const int r  = idx / (PIN / 4);
        const int k  = idx - r * (PIN / 4);
        const int gr = (row0 + r) & 1023;
        const int gc = (col0 + 4 * k) & 1023;
        *(v4f*)&s_in[r * PIN + 4 * k] = *(const v4f*)(src + (size_t)gr * 1024 + gc);
    }
    __syncthreads();

    // --- row stage: 5 row-tiles x 2 col-tiles x 2 filters = 20 wave tasks ---
    for (int task = wave; task < 20; task += 8) {
        const int ff = task & 1;
        const int nt = (task >> 1) & 1;
        const int mt = task >> 2;                   // 0..4
        const int rowbase = mt * 16;
        const float* fmb = &s_fm[ff * 640 + lane * 2];
        const float* ab  = &s_in[(rowbase + lm) * PIN + nt * 32 + 2 * lh];
        v8f acc = {};
        #pragma unroll
        for (int c = 0; c < 10; ++c) {
            v2f b = *(const v2f*)(fmb + c * 64);    // filter chunk (B layout)
            v2f a = *(const v2f*)(ab + c * 4);      // data chunk   (A layout)
            acc = __builtin_amdgcn_wmma_f32_16x16x4_f32(
                false, a, false, b, (short)0, acc, false, false);
        }
        float* dst = ff ? s_xh : s_xl;
        const int dcol = nt * 16 + lm;
        const int drow = rowbase + 8 * lh;          // D: vgpr r -> M = r + 8*lh
        #pragma unroll
        for (int r = 0; r < 8; ++r)
            dst[(drow + r) * PMI + dcol] = acc[r];
    }
    __syncthreads();

    // --- col stage: 2 row-tiles x 2 col-tiles x 4 subbands = 16 wave tasks ---
    for (int task = wave; task < 16; task += 8) {
        const int sb = task & 3;                    // 0:LL 1:LH 2:HL 3:HH
        const int nt = (task >> 2) & 1;
        const int it = (task >> 3) & 1;
        const int ff = sb & 1;
        const float* xi  = (sb & 2) ? s_xh : s_xl;
        const float* fmb = &s_fm[ff * 640 + lane * 2];
        const int coloff = nt * 16 + lm;
        const int rbase  = it * 32;                 // K window base (2*i0)
        v8f acc = {};
        #pragma unroll
        for (int c = 0; c < 10; ++c) {
            v2f a = *(const v2f*)(fmb + c * 64);    // filter chunk (A layout)
            const int rr = rbase + c * 4 + 2 * lh;
            v2f b;                                  // data chunk (B layout)
            b.x = xi[rr * PMI + coloff];
            b.y = xi[(rr + 1) * PMI + coloff];
            acc = __builtin_amdgcn_wmma_f32_16x16x4_f32(
                false, a, false, b, (short)0, acc, false, false);
        }
        const int oy = oy0 + it * 16 + 8 * lh;
        const int ox = ox0 + nt * 16 + lm;
        float* op = out + ((size_t)sb * 24 + img) * (512u * 512u) + ox;
        #pragma unroll
        for (int r = 0; r < 8; ++r)
            op[(size_t)(oy + r) * 512] = acc[r];
    }
}

extern "C" void kernel_launch(void* const* d_in, const int* in_sizes, int n_in,
                              void* d_out, int out_size, void* d_ws, size_t ws_size,
                              hipStream_t stream) {
    (void)in_sizes; (void)n_in; (void)out_size; (void)d_ws; (void)ws_size;
    const float* x = (const float*)d_in[0];
    float* out = (float*)d_out;
    dim3 grid(512 / 32, 512 / 32, 8 * 3);   // 16 x 16 x 24 workgroups
    wfb2dec_wmma_kernel<<<grid, dim3(256, 1, 1), 0, stream>>>(x, out);
}